// TypedLinear_80762565034484
// MI455X (gfx1250) — compile-verified
//
#include <hip/hip_runtime.h>

typedef float v2f __attribute__((ext_vector_type(2)));
typedef float v4f __attribute__((ext_vector_type(4)));
typedef float v8f __attribute__((ext_vector_type(8)));

#define NROWS 32768
#define IN_F 1024
#define OUT_F 1024
#define NUM_TYPES 8

#define BM 64
#define BN 128
#define BK 16
#define KSTR 20   // padded LDS row stride (floats): 80B -> 16B aligned float4, conflict-free b64

// ---------------- bucketing kernels ----------------
// ws layout (ints): [0..7] counts, [8..15] offsets, [16..23] counts2, rowidx at byte 128

__global__ void tl_init(int* __restrict__ ws) {
    int t = threadIdx.x;
    if (t < 24) ws[t] = 0;
}

__global__ void tl_hist(const int* __restrict__ types, int* __restrict__ counts, int n) {
    int i = blockIdx.x * blockDim.x + threadIdx.x;
    if (i < n) atomicAdd(&counts[types[i]], 1);
}

__global__ void tl_scan(int* __restrict__ ws) {
    if (threadIdx.x == 0) {
        int acc = 0;
        for (int t = 0; t < NUM_TYPES; ++t) { ws[8 + t] = acc; acc += ws[t]; }
    }
}

__global__ void tl_scatter(const int* __restrict__ types, int* __restrict__ ws,
                           int* __restrict__ rowidx, int n) {
    int i = blockIdx.x * blockDim.x + threadIdx.x;
    if (i < n) {
        int t = types[i];
        int pos = ws[8 + t] + atomicAdd(&ws[16 + t], 1);
        rowidx[pos] = i;
    }
}

// ---------------- grouped GEMM with f32 WMMA ----------------
// grid: x = row tiles (worst case 512), y = type (8), z = col tiles (8)
__global__ __launch_bounds__(256)
void tl_gemm(const float* __restrict__ x,
             const float* __restrict__ weight,
             const float* __restrict__ bias,
             const int* __restrict__ counts,
             const int* __restrict__ offsets,
             const int* __restrict__ rowidx,
             float* __restrict__ out) {
    const int type = blockIdx.y;
    const int cnt = counts[type];
    const int rowBase = blockIdx.x * BM;
    if (rowBase >= cnt) return;            // empty tile for this type
    const int off = offsets[type];
    const int colBase = blockIdx.z * BN;

    __shared__ float ldsX[2][BM * KSTR];   // X tile: 64 rows x 16 k (padded)
    __shared__ float ldsW[2][BN * KSTR];   // W tile: 128 cols x 16 k (padded)

    const int tid   = threadIdx.x;
    const int lane  = tid & 31;
    const int wave  = tid >> 5;            // 8 waves
    const int waveM = wave & 3;            // 4 waves along M (16 rows each)
    const int waveN = wave >> 2;           // 2 waves along N (64 cols each)
    const int hi    = lane >> 4;           // lane half selects K pair / M+8
    const int lanelo = lane & 15;

    // --- global load assignments ---
    // X: thread -> row xr (0..63), k quad xk; gathered row index, clamped for tail
    const int xr = tid >> 2;
    const int xk = (tid & 3) * 4;
    int rl = rowBase + xr; if (rl >= cnt) rl = cnt - 1;
    const int grow = rowidx[off + rl];
    const float* xsrc = x + (size_t)grow * IN_F + xk;

    // W: thread -> cols c0 and c0+64, same k quad xk
    const int c0 = tid >> 2;               // 0..63
    const float* wbase = weight + (size_t)type * OUT_F * IN_F + xk;
    const float* wsrc0 = wbase + (size_t)(colBase + c0)      * IN_F;
    const float* wsrc1 = wbase + (size_t)(colBase + c0 + 64) * IN_F;

    v8f acc[4] = {};                       // 16x64 per wave = 4 x (16x16) tiles

    // preload chunk 0 into LDS buffer 0
    v4f rx  = *(const v4f*)(xsrc);
    v4f rw0 = *(const v4f*)(wsrc0);
    v4f rw1 = *(const v4f*)(wsrc1);
    *(v4f*)(&ldsX[0][xr * KSTR + xk])        = rx;
    *(v4f*)(&ldsW[0][c0 * KSTR + xk])        = rw0;
    *(v4f*)(&ldsW[0][(c0 + 64) * KSTR + xk]) = rw1;
    __syncthreads();

    int cur = 0;
    #pragma unroll 1
    for (int k0 = 0; k0 < IN_F; k0 += BK) {
        const int nk = k0 + BK;
        v4f nx, nw0, nw1;
        if (nk < IN_F) {                   // issue next chunk's global loads early
            nx  = *(const v4f*)(xsrc  + nk);
            nw0 = *(const v4f*)(wsrc0 + nk);
            nw1 = *(const v4f*)(wsrc1 + nk);
        }

        // compute current chunk: 4 K-steps x 4 N-frags = 16 WMMAs per wave
        const float* Xb = &ldsX[cur][(waveM * 16 + lanelo) * KSTR + 2 * hi];
        const float* Wb = &ldsW[cur][(waveN * 64 + lanelo) * KSTR + 2 * hi];
        #pragma unroll
        for (int kk = 0; kk < BK; kk += 4) {
            v2f a = *(const v2f*)(Xb + kk);
            #pragma unroll
            for (int j = 0; j < 4; ++j) {
                v2f b = *(const v2f*)(Wb + j * 16 * KSTR + kk);
                acc[j] = __builtin_amdgcn_wmma_f32_16x16x4_f32(
                    false, a, false, b, (short)0, acc[j], false, false);
            }
        }

        if (nk < IN_F) {
            __syncthreads();
            const int nxt = cur ^ 1;
            *(v4f*)(&ldsX[nxt][xr * KSTR + xk])        = nx;
            *(v4f*)(&ldsW[nxt][c0 * KSTR + xk])        = nw0;
            *(v4f*)(&ldsW[nxt][(c0 + 64) * KSTR + xk]) = nw1;
            __syncthreads();
            cur = nxt;
        }
    }

    // epilogue: C[r + 8*hi][lanelo] in acc[j][r]; add bias, scatter rows back
    #pragma unroll
    for (int j = 0; j < 4; ++j) {
        const int col = colBase + waveN * 64 + j * 16 + lanelo;
        const float bj = bias[type * OUT_F + col];
        #pragma unroll
        for (int r = 0; r < 8; ++r) {
            const int rowl = rowBase + waveM * 16 + r + 8 * hi;
            if (rowl < cnt) {
                const int gr = rowidx[off + rowl];
                out[(size_t)gr * OUT_F + col] = acc[j][r] + bj;
            }
        }
    }
}

// ---------------- launch ----------------
extern "C" void kernel_launch(void* const* d_in, const int* in_sizes, int n_in,
                              void* d_out, int out_size, void* d_ws, size_t ws_size,
                              hipStream_t stream) {
    const float* x      = (const float*)d_in[0];
    const int*   types  = (const int*)d_in[1];
    const float* weight = (const float*)d_in[2];
    const float* bias   = (const float*)d_in[3];
    float*       out    = (float*)d_out;

    int* ws      = (int*)d_ws;
    int* counts  = ws;
    int* offsets = ws + 8;
    int* rowidx  = (int*)((char*)d_ws + 128);

    tl_init<<<1, 32, 0, stream>>>(ws);
    tl_hist<<<NROWS / 256, 256, 0, stream>>>(types, counts, NROWS);
    tl_scan<<<1, 32, 0, stream>>>(ws);
    tl_scatter<<<NROWS / 256, 256, 0, stream>>>(types, ws, rowidx, NROWS);

    dim3 grid(NROWS / BM, NUM_TYPES, OUT_F / BN);   // 512 x 8 x 8, empties exit fast
    tl_gemm<<<grid, 256, 0, stream>>>(x, weight, bias, counts, offsets, rowidx, out);
}